// PolicyDecoder_38328288149876
// MI455X (gfx1250) — compile-verified
//
#include <hip/hip_runtime.h>
#include <math.h>

// ---- problem constants (match reference) ----
#define E_DIM   128
#define CTX     384
#define BB      16
#define NSEQ    1024
#define D_INN   768
#define D_ST    16
#define DT_R    24
#define KCONV   4
#define TAU     1.0f
#define SINK_ITERS 5

typedef __attribute__((ext_vector_type(2))) float v2f;
typedef __attribute__((ext_vector_type(8))) float v8f;
typedef __attribute__((ext_vector_type(4))) unsigned int u32x4;
typedef __attribute__((ext_vector_type(4))) int i32x4;
typedef __attribute__((ext_vector_type(8))) int i32x8;

__device__ __forceinline__ float silu_f(float x) { return x / (1.0f + __expf(-x)); }
__device__ __forceinline__ float softplus_f(float x) { return (x > 20.0f) ? x : log1pf(__expf(x)); }

__device__ __forceinline__ v8f wmma_f32_16x16x4(v2f a, v2f b, v8f c) {
#if __has_builtin(__builtin_amdgcn_wmma_f32_16x16x4_f32)
  return __builtin_amdgcn_wmma_f32_16x16x4_f32(false, a, false, b, (short)0, c, false, false);
#else
  c[0] += a[0] * b[0];
  return c;
#endif
}

#if defined(__HIP_DEVICE_COMPILE__) && \
    __has_builtin(__builtin_amdgcn_tensor_load_to_lds) && \
    __has_builtin(__builtin_amdgcn_s_wait_tensorcnt)
#define HAVE_TDM 1
#else
#define HAVE_TDM 0
#endif

#define GT  64   // output tile edge (M and N)

#if HAVE_TDM
// ---- TDM path: K-chunk 32, double-buffered ----
#define KCH  32
#define LDP2 36  // 32 + 4 pad dwords (TDM pad_interval=32dw, pad_amount=4dw)

// Issue one TDM descriptor: 64 rows x 32 cols f32 tile (row stride ldElems)
// from global into LDS at ldsOff, hardware-padded to a 36-dword pitch.
__device__ __forceinline__ void tdm_load_tile(unsigned ldsOff, const float* gptr, int ldElems) {
  unsigned long long ga = (unsigned long long)(uintptr_t)gptr;
  u32x4 g0;
  g0[0] = 1u;                                        // count=1 valid user D#
  g0[1] = ldsOff;                                    // lds_addr (bytes)
  g0[2] = (unsigned)(ga & 0xFFFFFFFFu);              // global_addr[31:0]
  g0[3] = (unsigned)((ga >> 32) & 0x01FFFFFFu)       // global_addr[56:32]
        | (2u << 30);                                // type = 2 ("image")
  unsigned w0 = (2u << 16)                           // data_size = 4 bytes
              | (1u << 20)                           // pad_enable
              | (4u << 22)                           // pad_interval: 32 dwords
              | (3u << 25);                          // pad_amount: 4 dwords
  unsigned td0 = (unsigned)ldElems;                  // tensor_dim0 (elems/row)
  unsigned td1 = 0x40000000u;                        // rows: effectively unbounded
  unsigned long long s0 = (unsigned long long)ldElems; // tensor_dim0_stride
  i32x8 g1;
  g1[0] = (int)w0;                                   // wg_mask=0 (not in cluster)
  g1[1] = (int)((td0 & 0xFFFFu) << 16);              // tensor_dim0[15:0]
  g1[2] = (int)((td0 >> 16) | ((td1 & 0xFFFFu) << 16));
  g1[3] = (int)((td1 >> 16) | ((unsigned)KCH << 16)); // tile_dim0 = 32
  g1[4] = (int)GT;                                   // tile_dim1 = 64, tile_dim2 = 0
  g1[5] = (int)(s0 & 0xFFFFFFFFu);
  g1[6] = (int)((s0 >> 32) & 0xFFFFu);               // dim1 stride = 0 (2D)
  g1[7] = 0;
  i32x4 z4 = {0, 0, 0, 0};
#if __clang_major__ >= 23
  i32x8 z8 = {0, 0, 0, 0, 0, 0, 0, 0};
  __builtin_amdgcn_tensor_load_to_lds(g0, g1, z4, z4, z8, 0);
#else
  __builtin_amdgcn_tensor_load_to_lds(g0, g1, z4, z4, 0);
#endif
}
#else
#define LDP 68   // fallback path: 64 + 4 pad dwords
#endif

// ---------------------------------------------------------------------------
// Kernel 1: build ctx = [graph_emb | node_emb], RMSNorm -> h
// ---------------------------------------------------------------------------
__global__ __launch_bounds__(128)
void ctx_rms_kernel(const float* __restrict__ graph_emb,
                    const float* __restrict__ node_emb,
                    const float* __restrict__ rms_w,
                    float* __restrict__ ctx, float* __restrict__ h) {
  int row = blockIdx.x;
  int b = row >> 10;
  int t = threadIdx.x;
  float vals[3];
  float ss = 0.f;
#pragma unroll
  for (int i = 0; i < 3; ++i) {
    int c = t + i * 128;
    float x = (c < 2 * E_DIM) ? graph_emb[b * 2 * E_DIM + c]
                              : node_emb[(size_t)row * E_DIM + (c - 2 * E_DIM)];
    vals[i] = x;
    ss += x * x;
  }
  __shared__ float red[4];
  for (int o = 16; o > 0; o >>= 1) ss += __shfl_down(ss, o, 32);
  if ((t & 31) == 0) red[t >> 5] = ss;
  __syncthreads();
  float tot = red[0] + red[1] + red[2] + red[3];
  float inv = rsqrtf(tot / (float)CTX + 1e-6f);
#pragma unroll
  for (int i = 0; i < 3; ++i) {
    int c = t + i * 128;
    size_t o = (size_t)row * CTX + c;
    ctx[o] = vals[i];
    h[o]   = vals[i] * inv * rms_w[c];
  }
}

// ---------------------------------------------------------------------------
// WMMA f32 GEMM: C = scale * (A @ B[T]) + addScale * addSrc
// 64x64 output tile per 256-thread WG (8 waves x 16x32 strips).
// TDM path: double-buffered 64x32 K-chunks streamed by the Tensor Data Mover;
// wave 0 issues descriptors ahead and uses s_wait_tensorcnt 2 so chunk k's
// DMA (in-order per wave) completes while chunk k+1 is still in flight.
// ---------------------------------------------------------------------------
template <bool TRANSB, bool HASADD>
__global__ __launch_bounds__(256)
void wmma_gemm_kernel(const float* __restrict__ A, const float* __restrict__ Bsrc,
                      const float* __restrict__ addSrc, float* __restrict__ C,
                      int Kk, int lda, int ldb, int ldc,
                      long strideA, long strideB, long strideAdd, long strideC,
                      float scale, float addScale) {
  int tid = threadIdx.x;
  int n0 = blockIdx.x * GT;
  int m0 = blockIdx.y * GT;
  int batch = blockIdx.z;
  const float* Ab = A + (long)batch * strideA;
  const float* Bb = Bsrc + (long)batch * strideB;

  int wave = tid >> 5;
  int lane = tid & 31;
  int wm = wave >> 1;           // 0..3 : 16-row strip
  int wn = wave & 1;            // 0..1 : 32-col half
  int fr = lane & 15;           // fragment M (A) / N (B, D) index
  int kh = lane >> 4;           // K-half select (ISA 16x4 A layout)

  v8f acc0 = {};
  v8f acc1 = {};

#if HAVE_TDM
  __shared__ __align__(16) float lA[2][GT * LDP2];
  __shared__ __align__(16) float lB[2][GT * LDP2];

  auto issueChunk = [&](int bufi, int kb) {
    tdm_load_tile((unsigned)(uintptr_t)(void*)&lA[bufi][0],
                  Ab + (size_t)m0 * lda + kb, lda);
    if (TRANSB)
      tdm_load_tile((unsigned)(uintptr_t)(void*)&lB[bufi][0],
                    Bb + (size_t)n0 * ldb + kb, ldb);
    else
      tdm_load_tile((unsigned)(uintptr_t)(void*)&lB[bufi][0],
                    Bb + (size_t)kb * ldb + n0, ldb);
  };

  auto computeChunk = [&](int bufi) {
#pragma unroll
    for (int k4 = 0; k4 < KCH / 4; ++k4) {
      int kkb = 4 * k4 + 2 * kh;
      v2f a = *(const v2f*)&lA[bufi][(16 * wm + fr) * LDP2 + kkb];
      int nc0 = 32 * wn + fr;
      v2f b0, b1;
      if (TRANSB) {
        b0 = *(const v2f*)&lB[bufi][nc0 * LDP2 + kkb];
        b1 = *(const v2f*)&lB[bufi][(nc0 + 16) * LDP2 + kkb];
      } else {
        b0[0] = lB[bufi][kkb * LDP2 + nc0];
        b0[1] = lB[bufi][(kkb + 1) * LDP2 + nc0];
        b1[0] = lB[bufi][kkb * LDP2 + nc0 + 16];
        b1[1] = lB[bufi][(kkb + 1) * LDP2 + nc0 + 16];
      }
      acc0 = wmma_f32_16x16x4(a, b0, acc0);
      acc1 = wmma_f32_16x16x4(a, b1, acc1);
    }
  };

  int kChunks = Kk / KCH;       // all K here are multiples of 64 -> even
  if (wave == 0) issueChunk(0, 0);
  for (int kc = 0; kc < kChunks; kc += 2) {
    __syncthreads();            // buf1 consumers from previous iteration done
    if (wave == 0) {
      issueChunk(1, (kc + 1) * KCH);
      __builtin_amdgcn_s_wait_tensorcnt(2);   // chunk kc (buf0) landed
    }
    __syncthreads();
    computeChunk(0);
    __syncthreads();            // buf0 consumers done
    if (kc + 2 < kChunks) {
      if (wave == 0) {
        issueChunk(0, (kc + 2) * KCH);
        __builtin_amdgcn_s_wait_tensorcnt(2); // chunk kc+1 (buf1) landed
      }
    } else {
      if (wave == 0) __builtin_amdgcn_s_wait_tensorcnt(0);
    }
    __syncthreads();
    computeChunk(1);
  }
#else
  __shared__ __align__(16) float lA[GT * LDP];
  __shared__ __align__(16) float lB[GT * LDP];
  int kChunks = Kk / GT;
  for (int kc = 0; kc < kChunks; ++kc) {
    int kb = kc * GT;
#pragma unroll
    for (int i = 0; i < 4; ++i) {                 // 4 x float4 per thread
      int f = tid + 256 * i;
      int r = f >> 4, c4 = (f & 15) << 2;
      float4 va = *(const float4*)(Ab + (size_t)(m0 + r) * lda + kb + c4);
      *(float4*)&lA[r * LDP + c4] = va;
      float4 vb = TRANSB
          ? *(const float4*)(Bb + (size_t)(n0 + r) * ldb + kb + c4)
          : *(const float4*)(Bb + (size_t)(kb + r) * ldb + n0 + c4);
      *(float4*)&lB[r * LDP + c4] = vb;
    }
    if (kc + 1 < kChunks) {
      __builtin_prefetch(&Ab[(size_t)(m0 + (tid & 63)) * lda + kb + GT], 0, 1);
      __builtin_prefetch(TRANSB ? &Bb[(size_t)(n0 + (tid & 63)) * ldb + kb + GT]
                                : &Bb[(size_t)(kb + GT + (tid & 63)) * ldb + n0], 0, 1);
    }
    __syncthreads();
#pragma unroll
    for (int k4 = 0; k4 < 16; ++k4) {
      int kkb = 4 * k4 + 2 * kh;
      v2f a = *(const v2f*)&lA[(16 * wm + fr) * LDP + kkb];
      int nc0 = 32 * wn + fr;
      v2f b0, b1;
      if (TRANSB) {
        b0 = *(const v2f*)&lB[nc0 * LDP + kkb];
        b1 = *(const v2f*)&lB[(nc0 + 16) * LDP + kkb];
      } else {
        b0[0] = lB[kkb * LDP + nc0];      b0[1] = lB[(kkb + 1) * LDP + nc0];
        b1[0] = lB[kkb * LDP + nc0 + 16]; b1[1] = lB[(kkb + 1) * LDP + nc0 + 16];
      }
      acc0 = wmma_f32_16x16x4(a, b0, acc0);
      acc1 = wmma_f32_16x16x4(a, b1, acc1);
    }
    __syncthreads();
  }
#endif

  const float* addB = addSrc + (long)batch * strideAdd;
  float* Cb = C + (long)batch * strideC;
#pragma unroll
  for (int v = 0; v < 8; ++v) {
    int rowg = m0 + 16 * wm + v + 8 * kh;   // D layout: M = v (+8 for hi lanes)
    int colg = n0 + 32 * wn + fr;           // N = lane&15
    size_t o0 = (size_t)rowg * ldc + colg;
    size_t o1 = o0 + 16;
    float r0 = scale * acc0[v];
    float r1 = scale * acc1[v];
    if (HASADD) { r0 += addScale * addB[o0]; r1 += addScale * addB[o1]; }
    Cb[o0] = r0;
    Cb[o1] = r1;
  }
}

// ---------------------------------------------------------------------------
// causal depthwise conv (K=4) + SiLU on u (first half of uz rows)
// ---------------------------------------------------------------------------
__global__ __launch_bounds__(256)
void conv_silu_kernel(const float* __restrict__ uz,
                      const float* __restrict__ conv_w,
                      const float* __restrict__ conv_b,
                      float* __restrict__ uc) {
  size_t idx = (size_t)blockIdx.x * blockDim.x + threadIdx.x;
  if (idx >= (size_t)BB * NSEQ * D_INN) return;
  int d = (int)(idx % D_INN);
  size_t bn = idx / D_INN;
  int n = (int)(bn % NSEQ);
  size_t b = bn / NSEQ;
  float s = conv_b[d];
#pragma unroll
  for (int k = 0; k < KCONV; ++k) {
    int nn2 = n - (KCONV - 1) + k;
    if (nn2 >= 0)
      s += conv_w[d * KCONV + k] * uz[((b * NSEQ + nn2) * (size_t)(2 * D_INN)) + d];
  }
  uc[idx] = silu_f(s);
}

// ---------------------------------------------------------------------------
// dbc = u @ W_x (-> dt | B | C), delta = softplus(dt @ W_dt + b_dt)
// ---------------------------------------------------------------------------
__global__ __launch_bounds__(256)
void dbc_delta_kernel(const float* __restrict__ uc, const float* __restrict__ W_x,
                      const float* __restrict__ W_dt, const float* __restrict__ b_dt,
                      float* __restrict__ bc, float* __restrict__ delta) {
  int row = blockIdx.x;
  int t = threadIdx.x;
  __shared__ float lu[D_INN];
  __shared__ float ldt[DT_R];
  for (int i = t; i < D_INN; i += 256) lu[i] = uc[(size_t)row * D_INN + i];
  __syncthreads();
  if (t < DT_R + 2 * D_ST) {   // 56 outputs
    float s = 0.f;
    for (int k2 = 0; k2 < D_INN; ++k2) s += lu[k2] * W_x[(size_t)k2 * (DT_R + 2 * D_ST) + t];
    if (t < DT_R) ldt[t] = s;
    else bc[(size_t)row * (2 * D_ST) + (t - DT_R)] = s;
  }
  __syncthreads();
  for (int d = t; d < D_INN; d += 256) {
    float s = b_dt[d];
#pragma unroll
    for (int r = 0; r < DT_R; ++r) s += ldt[r] * W_dt[(size_t)r * D_INN + d];
    delta[(size_t)row * D_INN + d] = softplus_f(s);
  }
}

// ---------------------------------------------------------------------------
// S6 selective scan + Dp skip + SiLU(z) gate -> ybuf
// ---------------------------------------------------------------------------
#define SCH 32
__global__ __launch_bounds__(768)
void ssm_kernel(const float* __restrict__ uc, const float* __restrict__ uz,
                const float* __restrict__ bc, const float* __restrict__ delta,
                const float* __restrict__ A_log, const float* __restrict__ Dp,
                float* __restrict__ ybuf) {
  int b = blockIdx.x;
  int d = threadIdx.x;
  float Arow[D_ST];
#pragma unroll
  for (int s2 = 0; s2 < D_ST; ++s2) Arow[s2] = -__expf(A_log[d * D_ST + s2]);
  float hst[D_ST];
#pragma unroll
  for (int s2 = 0; s2 < D_ST; ++s2) hst[s2] = 0.f;
  float dp = Dp[d];
  __shared__ float lbc[SCH][2 * D_ST];
  for (int n0 = 0; n0 < NSEQ; n0 += SCH) {
    for (int i = d; i < SCH * 2 * D_ST; i += 768)
      lbc[i >> 5][i & 31] = bc[((size_t)b * NSEQ + n0 + (i >> 5)) * (2 * D_ST) + (i & 31)];
    __syncthreads();
    for (int nn = 0; nn < SCH; ++nn) {
      size_t o = (size_t)b * NSEQ + n0 + nn;
      float dlt = delta[o * D_INN + d];
      float ut  = uc[o * D_INN + d];
      float du  = dlt * ut;
      float y = 0.f;
#pragma unroll
      for (int s2 = 0; s2 < D_ST; ++s2) {
        float dA = __expf(dlt * Arow[s2]);
        hst[s2] = dA * hst[s2] + du * lbc[nn][s2];
        y += hst[s2] * lbc[nn][D_ST + s2];
      }
      float z = uz[o * (size_t)(2 * D_INN) + D_INN + d];
      ybuf[o * D_INN + d] = (y + dp * ut) * silu_f(z);
    }
    __syncthreads();
  }
}

// ---------------------------------------------------------------------------
// Sinkhorn with dual variables: la0 stays read-only (L2-resident, 67MB<192MB)
// ---------------------------------------------------------------------------
__global__ void zero_kernel(float* __restrict__ p, int n) {
  int i = blockIdx.x * 256 + threadIdx.x;
  if (i < n) p[i] = 0.f;
}

__global__ __launch_bounds__(256)
void col_lse_kernel(const float* __restrict__ la0, const float* __restrict__ uvec,
                    float* __restrict__ vvec) {
  int b = blockIdx.y;
  int j = blockIdx.x * 256 + threadIdx.x;
  const float* base = la0 + (size_t)b * NSEQ * NSEQ;
  const float* ub = uvec + b * NSEQ;
  float m = -1e30f, s = 0.f;
  for (int i = 0; i < NSEQ; ++i) {
    float x = base[(size_t)i * NSEQ + j] - ub[i];
    float nm = fmaxf(m, x);
    s = s * __expf(m - nm) + __expf(x - nm);
    m = nm;
  }
  vvec[b * NSEQ + j] = m + __logf(s);
}

__global__ __launch_bounds__(256)
void row_lse_kernel(const float* __restrict__ la0, const float* __restrict__ vvec,
                    float* __restrict__ uvec) {
  int row = blockIdx.x;
  int b = row >> 10;
  int i = row & 1023;
  int t = threadIdx.x;
  const float* base = la0 + (size_t)b * NSEQ * NSEQ + (size_t)i * NSEQ;
  const float* vb = vvec + b * NSEQ;
  float m = -1e30f, s = 0.f;
  for (int j = t; j < NSEQ; j += 256) {
    float x = base[j] - vb[j];
    float nm = fmaxf(m, x);
    s = s * __expf(m - nm) + __expf(x - nm);
    m = nm;
  }
  __shared__ float sm[256], ssum[256];
  sm[t] = m; ssum[t] = s;
  __syncthreads();
  for (int o = 128; o > 0; o >>= 1) {
    if (t < o) {
      float m2 = sm[t + o], s2 = ssum[t + o];
      float nm = fmaxf(sm[t], m2);
      ssum[t] = ssum[t] * __expf(sm[t] - nm) + s2 * __expf(m2 - nm);
      sm[t] = nm;
    }
    __syncthreads();
  }
  if (t == 0) uvec[row] = sm[0] + __logf(ssum[0]);
}

__global__ __launch_bounds__(256)
void final_kernel(const float* __restrict__ la0, const float* __restrict__ uvec,
                  const float* __restrict__ vvec, const float* __restrict__ gsamp,
                  int* __restrict__ tours, float* __restrict__ lp, float* __restrict__ ent) {
  int row = blockIdx.x;
  int b = row >> 10;
  int i = row & 1023;
  int t = threadIdx.x;
  const float* base = la0 + (size_t)b * NSEQ * NSEQ + (size_t)i * NSEQ;
  const float* gb   = gsamp + (size_t)b * NSEQ * NSEQ + (size_t)i * NSEQ;
  const float* vb = vvec + b * NSEQ;
  float ui = uvec[row];
  float bestKey = -1e30f; int bestJ = 0; float bestLa = 0.f; float es = 0.f;
  for (int j = t; j < NSEQ; j += 256) {
    float la = base[j] - ui - vb[j];
    float key = la + gb[j];
    if (key > bestKey) { bestKey = key; bestJ = j; bestLa = la; }
    es += la * __expf(la);
  }
  __shared__ float sk[256]; __shared__ int sj[256];
  __shared__ float sl[256]; __shared__ float se[256];
  sk[t] = bestKey; sj[t] = bestJ; sl[t] = bestLa; se[t] = es;
  __syncthreads();
  for (int o = 128; o > 0; o >>= 1) {
    if (t < o) {
      if (sk[t + o] > sk[t]) { sk[t] = sk[t + o]; sj[t] = sj[t + o]; sl[t] = sl[t + o]; }
      se[t] += se[t + o];
    }
    __syncthreads();
  }
  if (t == 0) { tours[row] = sj[0]; lp[row] = sl[0]; ent[row] = -se[0]; }
}

// ---------------------------------------------------------------------------
extern "C" void kernel_launch(void* const* d_in, const int* in_sizes, int n_in,
                              void* d_out, int out_size, void* d_ws, size_t ws_size,
                              hipStream_t stream) {
  (void)in_sizes; (void)n_in; (void)out_size; (void)ws_size;
  const float* graph_emb = (const float*)d_in[0];
  const float* node_emb  = (const float*)d_in[1];
  const float* W_key     = (const float*)d_in[2];
  const float* rms_w     = (const float*)d_in[3];
  const float* W_in      = (const float*)d_in[4];
  const float* conv_w    = (const float*)d_in[5];
  const float* conv_b    = (const float*)d_in[6];
  const float* W_x       = (const float*)d_in[7];
  const float* W_dt      = (const float*)d_in[8];
  const float* b_dt      = (const float*)d_in[9];
  const float* A_log     = (const float*)d_in[10];
  const float* Dp        = (const float*)d_in[11];
  const float* W_out     = (const float*)d_in[12];
  const float* gsink     = (const float*)d_in[13];
  const float* gsamp     = (const float*)d_in[14];

  float* ws = (float*)d_ws;
  const size_t M = (size_t)BB * NSEQ;           // 16384
  float* ctx   = ws;
  float* h     = ctx   + M * CTX;
  float* uz    = h     + M * CTX;
  float* uc    = uz    + M * 2 * D_INN;
  float* bc    = uc    + M * D_INN;
  float* delta = bc    + M * 2 * D_ST;
  float* ybuf  = delta + M * D_INN;
  float* keys  = ybuf  + M * D_INN;
  float* qrs   = keys  + M * CTX;
  float* la0   = qrs   + M * CTX;
  float* uvec  = la0   + (size_t)BB * NSEQ * NSEQ;
  float* vvec  = uvec  + M;

  // 1. ctx + RMSNorm
  ctx_rms_kernel<<<(int)M, 128, 0, stream>>>(graph_emb, node_emb, rms_w, ctx, h);

  // 2. keys = node_emb @ W_key   (16384 x 384, K=128)
  {
    dim3 g(CTX / GT, (int)M / GT, 1);
    wmma_gemm_kernel<false, false><<<g, 256, 0, stream>>>(node_emb, W_key, nullptr, keys,
        E_DIM, E_DIM, CTX, CTX, 0, 0, 0, 0, 1.f, 0.f);
  }
  // 3. uz = h @ W_in             (16384 x 1536, K=384)
  {
    dim3 g((2 * D_INN) / GT, (int)M / GT, 1);
    wmma_gemm_kernel<false, false><<<g, 256, 0, stream>>>(h, W_in, nullptr, uz,
        CTX, CTX, 2 * D_INN, 2 * D_INN, 0, 0, 0, 0, 1.f, 0.f);
  }
  // 4. causal depthwise conv + SiLU
  {
    size_t tot = M * D_INN;
    conv_silu_kernel<<<(int)((tot + 255) / 256), 256, 0, stream>>>(uz, conv_w, conv_b, uc);
  }
  // 5. dbc / delta
  dbc_delta_kernel<<<(int)M, 256, 0, stream>>>(uc, W_x, W_dt, b_dt, bc, delta);
  // 6. selective scan + gate
  ssm_kernel<<<BB, D_INN, 0, stream>>>(uc, uz, bc, delta, A_log, Dp, ybuf);
  // 7. queries = ctx + ybuf @ W_out
  {
    dim3 g(CTX / GT, (int)M / GT, 1);
    wmma_gemm_kernel<false, true><<<g, 256, 0, stream>>>(ybuf, W_out, ctx, qrs,
        D_INN, D_INN, CTX, CTX, 0, 0, 0, 0, 1.f, 1.f);
  }
  // 8. la0 = (keys @ queries^T + gumbel_sink) / tau   (batched, B^T)
  {
    dim3 g(NSEQ / GT, NSEQ / GT, BB);
    wmma_gemm_kernel<true, true><<<g, 256, 0, stream>>>(keys, qrs, gsink, la0,
        CTX, CTX, CTX, NSEQ,
        (long)NSEQ * CTX, (long)NSEQ * CTX, (long)NSEQ * NSEQ, (long)NSEQ * NSEQ,
        1.0f / TAU, 1.0f / TAU);
  }
  // 9. Sinkhorn with dual variables u,v (la0 read-only, lives in L2)
  zero_kernel<<<(int)((M + 255) / 256), 256, 0, stream>>>(uvec, (int)M);
  for (int it = 0; it < SINK_ITERS; ++it) {
    dim3 gc(NSEQ / 256, BB);
    col_lse_kernel<<<gc, 256, 0, stream>>>(la0, uvec, vvec);
    row_lse_kernel<<<(int)M, 256, 0, stream>>>(la0, vvec, uvec);
  }
  // 10. outputs: tours (int), log_probs, entropies
  int*   tours = (int*)d_out;
  float* lp    = (float*)d_out + M;
  float* ent   = (float*)d_out + 2 * M;
  final_kernel<<<(int)M, 256, 0, stream>>>(la0, uvec, vvec, gsamp, tours, lp, ent);
}